// SigmoidGHMLoss_59777354826345
// MI455X (gfx1250) — compile-verified
//
#include <hip/hip_runtime.h>
#include <stdint.h>

#define BINS 10
#define THREADS 256

typedef float v4f __attribute__((ext_vector_type(4)));

// idx = clip(floor(|sigmoid(x) - t| * 10), 0, 9)
__device__ __forceinline__ int bin_of(float x, float t) {
  float e = __expf(-x);                         // v_exp_f32
  float p = __builtin_amdgcn_rcpf(1.0f + e);    // v_rcp_f32 (sigmoid)
  float g = fabsf(p - t);                       // in [0,1]
  int b = (int)(g * (float)BINS);               // g >= 0 -> trunc == floor
  b = b > (BINS - 1) ? (BINS - 1) : b;
  return b < 0 ? 0 : b;
}

// stable BCE-with-logits: max(x,0) - x*t + log1p(exp(-|x|))
__device__ __forceinline__ float bce_of(float x, float t) {
  float l = __logf(1.0f + __expf(-fabsf(x)));   // arg in (1,2]: accurate
  return fmaxf(x, 0.0f) - x * t + l;
}

__global__ void ghm_init(unsigned* __restrict__ counts) {
  if (threadIdx.x < 16) counts[threadIdx.x] = 0u;
}

__global__ void ghm_hist(const float* __restrict__ in, const float* __restrict__ tg,
                         unsigned* __restrict__ counts, int n) {
  __shared__ unsigned h[BINS];
  if (threadIdx.x < BINS) h[threadIdx.x] = 0u;
  __syncthreads();

  const int n4 = n >> 2;
  const int gtid = blockIdx.x * blockDim.x + threadIdx.x;
  const int stride = gridDim.x * blockDim.x;

  // ten 6-bit fields packed in one u64: pack += 1 << (6*idx); flush every 15
  // float4 iterations (max 60 increments per field < 63 -> no overflow).
  unsigned wcnt[BINS];
#pragma unroll
  for (int b = 0; b < BINS; ++b) wcnt[b] = 0u;
  unsigned long long pack = 0ull;
  int budget = 15;

  const v4f* in4 = (const v4f*)in;
  const v4f* tg4 = (const v4f*)tg;

  for (int i = gtid; i < n4; i += stride) {
    __builtin_prefetch((const void*)(in4 + i + stride), 0, 0); // global_prefetch_b8
    __builtin_prefetch((const void*)(tg4 + i + stride), 0, 0);
    v4f x = __builtin_nontemporal_load(in4 + i);               // b128, TH=NT
    v4f t = __builtin_nontemporal_load(tg4 + i);
    pack += (1ull << (6 * bin_of(x.x, t.x)))
          + (1ull << (6 * bin_of(x.y, t.y)))
          + (1ull << (6 * bin_of(x.z, t.z)))
          + (1ull << (6 * bin_of(x.w, t.w)));
    if (--budget == 0) {
#pragma unroll
      for (int b = 0; b < BINS; ++b) wcnt[b] += (unsigned)(pack >> (6 * b)) & 63u;
      pack = 0ull;
      budget = 15;
    }
  }
#pragma unroll
  for (int b = 0; b < BINS; ++b) wcnt[b] += (unsigned)(pack >> (6 * b)) & 63u;

  // scalar tail (n not multiple of 4)
  for (int i = (n4 << 2) + gtid; i < n; i += stride) {
    atomicAdd(&h[bin_of(in[i], tg[i])], 1u);
  }

  // per-lane -> block histogram (bins land in distinct LDS banks)
#pragma unroll
  for (int b = 0; b < BINS; ++b)
    if (wcnt[b]) atomicAdd(&h[b], wcnt[b]);
  __syncthreads();
  if (threadIdx.x < BINS && h[threadIdx.x]) atomicAdd(&counts[threadIdx.x], h[threadIdx.x]);
}

// lut[b] = (counts[b] > 0) ? ((tot / max(counts,1)) / max(n_nonempty,1)) / tot : 0
__global__ void ghm_finalize(const unsigned* __restrict__ counts,
                             float* __restrict__ lut, int n) {
  if (threadIdx.x == 0) {
    int nb = 0;
#pragma unroll
    for (int b = 0; b < BINS; ++b) nb += (counts[b] > 0u) ? 1 : 0;
    float nf = fmaxf((float)nb, 1.0f);
    float tot = (float)n;
    float invtot = 1.0f / tot;
#pragma unroll
    for (int b = 0; b < BINS; ++b) {
      float c = (float)counts[b];
      lut[b] = (counts[b] > 0u) ? ((tot / fmaxf(c, 1.0f)) / nf) * invtot : 0.0f;
    }
  }
}

__global__ void ghm_out(const float* __restrict__ in, const float* __restrict__ tg,
                        const float* __restrict__ lut, float* __restrict__ out, int n) {
  __shared__ float slut[BINS];
  // double-buffered async staging: per-thread private 16B slots
  __shared__ v4f sx[2][THREADS];
  __shared__ v4f st[2][THREADS];
  if (threadIdx.x < BINS) slut[threadIdx.x] = lut[threadIdx.x];
  __syncthreads();

  const int n4 = n >> 2;
  const int gtid = blockIdx.x * blockDim.x + threadIdx.x;
  const int stride = gridDim.x * blockDim.x;
  const v4f* in4 = (const v4f*)in;
  const v4f* tg4 = (const v4f*)tg;
  v4f* out4 = (v4f*)out;

  // issue one async global->LDS b128 pair for tile index i_ into buffer b_
  auto issue = [&](int i_, int b_) {
    int ii = i_ < n4 ? i_ : n4 - 1;                    // clamp: always issue (uniform EXEC)
    unsigned lx = (unsigned)(uintptr_t)&sx[b_][threadIdx.x];
    unsigned lt = (unsigned)(uintptr_t)&st[b_][threadIdx.x];
    unsigned long long gx = (unsigned long long)(in4 + ii);
    unsigned long long gt = (unsigned long long)(tg4 + ii);
    asm volatile("global_load_async_to_lds_b128 %0, %1, off th:TH_LOAD_NT"
                 :: "v"(lx), "v"(gx) : "memory");
    asm volatile("global_load_async_to_lds_b128 %0, %1, off th:TH_LOAD_NT"
                 :: "v"(lt), "v"(gt) : "memory");
  };

  const int niter = (n4 + stride - 1) / stride;        // uniform across grid
  if (n4 > 0) {
    int buf = 0;
    issue(gtid, 0);
    for (int k = 0; k < niter; ++k) {
      const int i = gtid + k * stride;
      if (k + 1 < niter) {                             // uniform branch
        issue(gtid + (k + 1) * stride, buf ^ 1);
        asm volatile("s_wait_asynccnt 0x2" ::: "memory");  // pair k has landed
      } else {
        asm volatile("s_wait_asynccnt 0x0" ::: "memory");
      }
      v4f x = sx[buf][threadIdx.x];                    // ds_load_b128
      v4f t = st[buf][threadIdx.x];
      if (i < n4) {
        v4f o;
        o.x = slut[bin_of(x.x, t.x)] * bce_of(x.x, t.x);
        o.y = slut[bin_of(x.y, t.y)] * bce_of(x.y, t.y);
        o.z = slut[bin_of(x.z, t.z)] * bce_of(x.z, t.z);
        o.w = slut[bin_of(x.w, t.w)] * bce_of(x.w, t.w);
        __builtin_nontemporal_store(o, out4 + i);
      }
      buf ^= 1;
    }
  }

  // scalar tail (n not multiple of 4): direct loads
  for (int i = (n4 << 2) + gtid; i < n; i += stride) {
    out[i] = slut[bin_of(in[i], tg[i])] * bce_of(in[i], tg[i]);
  }
}

extern "C" void kernel_launch(void* const* d_in, const int* in_sizes, int n_in,
                              void* d_out, int out_size, void* d_ws, size_t ws_size,
                              hipStream_t stream) {
  const float* in = (const float*)d_in[0];
  const float* tg = (const float*)d_in[1];
  float* out = (float*)d_out;
  const int n = in_sizes[0];   // 8*80000*80 = 51,200,000

  unsigned* counts = (unsigned*)d_ws;               // 10 (padded 16) u32 counters
  float* lut = (float*)((char*)d_ws + 64);          // 10 float weights

  ghm_init<<<1, 32, 0, stream>>>(counts);

  const int blocks = 4096;   // ~32K wave32: enough MLP to saturate 23.3 TB/s
  ghm_hist<<<blocks, THREADS, 0, stream>>>(in, tg, counts, n);
  ghm_finalize<<<1, 32, 0, stream>>>(counts, lut, n);
  ghm_out<<<blocks, THREADS, 0, stream>>>(in, tg, lut, out, n);
}